// TreeDecoder_17025250361512
// MI455X (gfx1250) — compile-verified
//
#include <hip/hip_runtime.h>
#include <math.h>

#define B_ 128
#define N_ 512
#define H_ 512
#define L_ 256
#define P_ 3
#define T_ 511
#define HS 516   // padded LDS row stride (floats); 16B-aligned rows

typedef float v2f __attribute__((ext_vector_type(2)));
typedef float v4f __attribute__((ext_vector_type(4)));
typedef float v8f __attribute__((ext_vector_type(8)));

__device__ __forceinline__ v8f wmma4(v2f a, v2f b, v8f c) {
  // D = A(16x4 f32) * B(4x16 f32) + C(16x16 f32)
  return __builtin_amdgcn_wmma_f32_16x16x4_f32(false, a, false, b, (short)0, c,
                                               false, false);
}
__device__ __forceinline__ v2f lo2(v4f x) { return __builtin_shufflevector(x, x, 0, 1); }
__device__ __forceinline__ v2f hi2(v4f x) { return __builtin_shufflevector(x, x, 2, 3); }

__device__ __forceinline__ float sigmoidf_(float x) {
  return 1.0f / (1.0f + expf(-x));
}
__device__ __forceinline__ float softplusf_(float x) {
  return fmaxf(x, 0.0f) + log1pf(expf(-fabsf(x)));
}

// ---------------------------------------------------------------------------
// embeds[b,n,h] = b_ne[h] + sum_p feat[b,n,p] * W_ne[h,p]   (K=4, feat[3]=0)
// ---------------------------------------------------------------------------
__global__ void embed_kernel(const float* __restrict__ tpos,
                             const float* __restrict__ Wne,
                             const float* __restrict__ bne,
                             float* __restrict__ emb) {
  size_t idx = (size_t)blockIdx.x * blockDim.x + threadIdx.x;
  size_t total = (size_t)B_ * N_ * H_;
  if (idx >= total) return;
  int h = (int)(idx % H_);
  size_t bn = idx / H_;
  int n = (int)(bn % N_);
  int b = (int)(bn / N_);
  float p0 = 0.f, p1 = 0.f, p2 = 0.f;
  if (n != 0) {  // pos_full[:,0,:] = 0
    const float* q = tpos + ((size_t)b * N_ + n) * P_;
    p0 = q[0]; p1 = q[1]; p2 = q[2];
  }
  const float* w = Wne + (size_t)h * 4;  // W_ne is (H, P+1)
  emb[idx] = bne[h] + p0 * w[0] + p1 * w[1] + p2 * w[2];
}

// ---------------------------------------------------------------------------
// Generic fp32 WMMA GEMM: Out[m, n] = act( A[m,:] . W[n,:] + bias[n] )
//   A: rows addressed via rowmap (0: direct, 1: bt -> embeds row b*N + t)
//   W: (Nout, K) row-major (computes A @ W^T)
// One wave = 16x64 output block (4 N-tiles sharing the A operand).
// K-permuted macro-steps of 8 -> b128 operand loads; depth-2 pipeline.
// K must be a multiple of 16.
// ---------------------------------------------------------------------------
__global__ __launch_bounds__(128)
__attribute__((amdgpu_waves_per_eu(2, 8)))
void gemm_wmma(
    const float* __restrict__ A, long long lda,
    const float* __restrict__ W, const float* __restrict__ bias,
    float* __restrict__ Out, long long ldo,
    int M, int Nout, int K, int rowmap, int act) {
  const int lane = threadIdx.x & 31;
  const int lr = lane & 15;
  const int half = lane >> 4;
  const int ngrp = Nout >> 6;
  const int mtiles = M >> 4;
  int gw = blockIdx.x * (blockDim.x >> 5) + (threadIdx.x >> 5);
  if (gw >= mtiles * ngrp) return;  // wave-uniform
  const int mt = gw / ngrp;
  const int ng = gw - mt * ngrp;

  const int mrow = mt * 16 + lr;  // A row loaded by this lane
  long long arow;
  if (rowmap == 1) {
    int bb = mrow / T_;
    int tt = mrow - bb * T_;
    arow = (long long)bb * N_ + tt;
  } else {
    arow = mrow;
  }
  // half 0 covers physical k0..k0+3, half 1 covers k0+4..k0+7 (perm shared A&B)
  const float* Ap = A + arow * lda + 4 * half;
  const int n0 = ng * 64;
  const float* Wp = W + (size_t)(n0 + lr) * K + 4 * half;

  v8f acc0 = {0,0,0,0,0,0,0,0}, acc1 = {0,0,0,0,0,0,0,0};
  v8f acc2 = {0,0,0,0,0,0,0,0}, acc3 = {0,0,0,0,0,0,0,0};
  const int S = K >> 3;  // >= 2
  // stage A (=s) and stage B (=s+1) preloads
  v4f aA  = *(const v4f*)(Ap);
  v4f b0A = *(const v4f*)(Wp);
  v4f b1A = *(const v4f*)(Wp + (size_t)16 * K);
  v4f b2A = *(const v4f*)(Wp + (size_t)32 * K);
  v4f b3A = *(const v4f*)(Wp + (size_t)48 * K);
  v4f aB  = *(const v4f*)(Ap + 8);
  v4f b0B = *(const v4f*)(Wp + 8);
  v4f b1B = *(const v4f*)(Wp + (size_t)16 * K + 8);
  v4f b2B = *(const v4f*)(Wp + (size_t)32 * K + 8);
  v4f b3B = *(const v4f*)(Wp + (size_t)48 * K + 8);
  for (int s = 0; s < S - 2; ++s) {
    const int kn = (s + 2) * 8;  // prefetch stage s+2
    v4f aN  = *(const v4f*)(Ap + kn);
    v4f b0N = *(const v4f*)(Wp + kn);
    v4f b1N = *(const v4f*)(Wp + (size_t)16 * K + kn);
    v4f b2N = *(const v4f*)(Wp + (size_t)32 * K + kn);
    v4f b3N = *(const v4f*)(Wp + (size_t)48 * K + kn);
    acc0 = wmma4(lo2(aA), lo2(b0A), acc0);
    acc1 = wmma4(lo2(aA), lo2(b1A), acc1);
    acc2 = wmma4(lo2(aA), lo2(b2A), acc2);
    acc3 = wmma4(lo2(aA), lo2(b3A), acc3);
    acc0 = wmma4(hi2(aA), hi2(b0A), acc0);
    acc1 = wmma4(hi2(aA), hi2(b1A), acc1);
    acc2 = wmma4(hi2(aA), hi2(b2A), acc2);
    acc3 = wmma4(hi2(aA), hi2(b3A), acc3);
    aA = aB; b0A = b0B; b1A = b1B; b2A = b2B; b3A = b3B;
    aB = aN; b0B = b0N; b1B = b1N; b2B = b2N; b3B = b3N;
  }
  // stage S-2
  acc0 = wmma4(lo2(aA), lo2(b0A), acc0);
  acc1 = wmma4(lo2(aA), lo2(b1A), acc1);
  acc2 = wmma4(lo2(aA), lo2(b2A), acc2);
  acc3 = wmma4(lo2(aA), lo2(b3A), acc3);
  acc0 = wmma4(hi2(aA), hi2(b0A), acc0);
  acc1 = wmma4(hi2(aA), hi2(b1A), acc1);
  acc2 = wmma4(hi2(aA), hi2(b2A), acc2);
  acc3 = wmma4(hi2(aA), hi2(b3A), acc3);
  // stage S-1
  acc0 = wmma4(lo2(aB), lo2(b0B), acc0);
  acc1 = wmma4(lo2(aB), lo2(b1B), acc1);
  acc2 = wmma4(lo2(aB), lo2(b2B), acc2);
  acc3 = wmma4(lo2(aB), lo2(b3B), acc3);
  acc0 = wmma4(hi2(aB), hi2(b0B), acc0);
  acc1 = wmma4(hi2(aB), hi2(b1B), acc1);
  acc2 = wmma4(hi2(aB), hi2(b2B), acc2);
  acc3 = wmma4(hi2(aB), hi2(b3B), acc3);

#pragma unroll
  for (int v = 0; v < 8; ++v) {
    size_t orow = (size_t)(mt * 16 + v + 8 * half);
#pragma unroll
    for (int j = 0; j < 4; ++j) {
      int col = n0 + j * 16 + lr;
      float s = (j == 0 ? acc0[v] : j == 1 ? acc1[v] : j == 2 ? acc2[v] : acc3[v]);
      s += bias[col];
      if (act == 1) s = tanhf(s);
      Out[orow * ldo + col] = s;
    }
  }
}

// ---------------------------------------------------------------------------
// Persistent GRU: 8 blocks x 16 batch rows, 16 waves x (2 groups of 16 cols).
// Each wave computes its own r/u/n hp tiles via f32 WMMA, h lives in LDS.
// b128 operand loads (LDS A, global W_hh) + software pipelining.
// ---------------------------------------------------------------------------
__global__ __launch_bounds__(512) void gru_kernel(
    const float* __restrict__ h0g, const float* __restrict__ Xp,
    const float* __restrict__ Whh, const float* __restrict__ bhh,
    float* __restrict__ Hs) {
  __shared__ float hsm[16 * HS];
  const int tid = threadIdx.x;
  const int wave = tid >> 5;
  const int lane = tid & 31;
  const int lr = lane & 15;
  const int half = lane >> 4;
  const int b0 = blockIdx.x * 16;

  for (int idx = tid; idx < 16 * H_; idx += 512) {
    int row = idx >> 9, col = idx & (H_ - 1);
    hsm[row * HS + col] = h0g[(size_t)(b0 + row) * H_ + col];
  }
  __syncthreads();

  const int c1 = wave * 16;
  const int c2 = c1 + 256;
  // W_hh rows: gate r -> [0,512), u -> [512,1024), n -> [1024,1536)
  const float* W1r = Whh + (size_t)(c1 + lr) * H_ + 4 * half;
  const float* W1u = Whh + (size_t)(512 + c1 + lr) * H_ + 4 * half;
  const float* W1n = Whh + (size_t)(1024 + c1 + lr) * H_ + 4 * half;
  const float* W2r = Whh + (size_t)(c2 + lr) * H_ + 4 * half;
  const float* W2u = Whh + (size_t)(512 + c2 + lr) * H_ + 4 * half;
  const float* W2n = Whh + (size_t)(1024 + c2 + lr) * H_ + 4 * half;
  const float bh1r = bhh[c1 + lr], bh1u = bhh[512 + c1 + lr], bh1n = bhh[1024 + c1 + lr];
  const float bh2r = bhh[c2 + lr], bh2u = bhh[512 + c2 + lr], bh2n = bhh[1024 + c2 + lr];
  const float* hp_ = &hsm[lr * HS + 4 * half];

  for (int t = 0; t < T_; ++t) {
    v8f r1 = {0,0,0,0,0,0,0,0}, u1 = {0,0,0,0,0,0,0,0}, n1 = {0,0,0,0,0,0,0,0};
    v8f r2 = {0,0,0,0,0,0,0,0}, u2 = {0,0,0,0,0,0,0,0}, n2 = {0,0,0,0,0,0,0,0};
    v4f aC   = *(const v4f*)(hp_);
    v4f r1C  = *(const v4f*)(W1r);
    v4f u1C  = *(const v4f*)(W1u);
    v4f n1C  = *(const v4f*)(W1n);
    v4f r2C  = *(const v4f*)(W2r);
    v4f u2C  = *(const v4f*)(W2u);
    v4f n2C  = *(const v4f*)(W2n);
    for (int s = 0; s < (H_ / 8) - 1; ++s) {
      const int kn = (s + 1) * 8;
      v4f aN  = *(const v4f*)(hp_ + kn);
      v4f r1N = *(const v4f*)(W1r + kn);
      v4f u1N = *(const v4f*)(W1u + kn);
      v4f n1N = *(const v4f*)(W1n + kn);
      v4f r2N = *(const v4f*)(W2r + kn);
      v4f u2N = *(const v4f*)(W2u + kn);
      v4f n2N = *(const v4f*)(W2n + kn);
      r1 = wmma4(lo2(aC), lo2(r1C), r1);
      u1 = wmma4(lo2(aC), lo2(u1C), u1);
      n1 = wmma4(lo2(aC), lo2(n1C), n1);
      r2 = wmma4(lo2(aC), lo2(r2C), r2);
      u2 = wmma4(lo2(aC), lo2(u2C), u2);
      n2 = wmma4(lo2(aC), lo2(n2C), n2);
      r1 = wmma4(hi2(aC), hi2(r1C), r1);
      u1 = wmma4(hi2(aC), hi2(u1C), u1);
      n1 = wmma4(hi2(aC), hi2(n1C), n1);
      r2 = wmma4(hi2(aC), hi2(r2C), r2);
      u2 = wmma4(hi2(aC), hi2(u2C), u2);
      n2 = wmma4(hi2(aC), hi2(n2C), n2);
      aC = aN; r1C = r1N; u1C = u1N; n1C = n1N; r2C = r2N; u2C = u2N; n2C = n2N;
    }
    r1 = wmma4(lo2(aC), lo2(r1C), r1);
    u1 = wmma4(lo2(aC), lo2(u1C), u1);
    n1 = wmma4(lo2(aC), lo2(n1C), n1);
    r2 = wmma4(lo2(aC), lo2(r2C), r2);
    u2 = wmma4(lo2(aC), lo2(u2C), u2);
    n2 = wmma4(lo2(aC), lo2(n2C), n2);
    r1 = wmma4(hi2(aC), hi2(r1C), r1);
    u1 = wmma4(hi2(aC), hi2(u1C), u1);
    n1 = wmma4(hi2(aC), hi2(n1C), n1);
    r2 = wmma4(hi2(aC), hi2(r2C), r2);
    u2 = wmma4(hi2(aC), hi2(u2C), u2);
    n2 = wmma4(hi2(aC), hi2(n2C), n2);

    float h2a[8], h2b[8];
#pragma unroll
    for (int v = 0; v < 8; ++v) {
      int row = v + 8 * half;
      size_t xb = ((size_t)(b0 + row) * T_ + t) * (3 * H_);
      {  // column group 1
        int col = c1 + lr;
        float xr = Xp[xb + col], xu = Xp[xb + 512 + col], xn = Xp[xb + 1024 + col];
        float hr = r1[v] + bh1r, hu = u1[v] + bh1u, hn = n1[v] + bh1n;
        float rg = sigmoidf_(xr + hr);
        float ug = sigmoidf_(xu + hu);
        float ng = tanhf(xn + rg * hn);
        float hp = hsm[row * HS + col];
        h2a[v] = (1.0f - ug) * ng + ug * hp;
      }
      {  // column group 2
        int col = c2 + lr;
        float xr = Xp[xb + col], xu = Xp[xb + 512 + col], xn = Xp[xb + 1024 + col];
        float hr = r2[v] + bh2r, hu = u2[v] + bh2u, hn = n2[v] + bh2n;
        float rg = sigmoidf_(xr + hr);
        float ug = sigmoidf_(xu + hu);
        float ng = tanhf(xn + rg * hn);
        float hp = hsm[row * HS + col];
        h2b[v] = (1.0f - ug) * ng + ug * hp;
      }
    }
    __syncthreads();  // all waves done reading h
#pragma unroll
    for (int v = 0; v < 8; ++v) {
      int row = v + 8 * half;
      hsm[row * HS + c1 + lr] = h2a[v];
      hsm[row * HS + c2 + lr] = h2b[v];
      size_t hb = ((size_t)(b0 + row) * T_ + t) * H_;
      Hs[hb + c1 + lr] = h2a[v];
      Hs[hb + c2 + lr] = h2b[v];
    }
    __syncthreads();  // h updated before next step reads it
  }
}

// ---------------------------------------------------------------------------
// Fused attention: per (b, 16-row t-tile):
//   Q tile = Hs tile @ W_q^T + b_q (WMMA, scaled 1/sqrt(H)) -> LDS
//   logits = Q tile @ K[b]^T       (WMMA) -> LDS (reuses Q buffer)
//   masked log-softmax + CE gather -> atomic add to loss_parent
// ---------------------------------------------------------------------------
__global__ __launch_bounds__(256)
__attribute__((amdgpu_waves_per_eu(2, 8)))
void attn_kernel(
    const float* __restrict__ Hsg, const float* __restrict__ Wq,
    const float* __restrict__ bq, const float* __restrict__ Kmat,
    const int* __restrict__ parent, float* __restrict__ out) {
  __shared__ float qlds[16 * HS];
  const int b = blockIdx.x >> 5;   // 32 t-tiles per batch row
  const int tt = blockIdx.x & 31;
  const int t0 = tt * 16;
  const int tid = threadIdx.x;
  const int wave = tid >> 5;
  const int lane = tid & 31;
  const int lr = lane & 15;
  const int half = lane >> 4;
  const float scale = 0.044194173824159216f;  // 1/sqrt(512)
  const int n0 = wave * 64;

  // phase 0: Q tile (rows t0..t0+15, clamped; invalid rows masked later)
  {
    int tcl = min(t0 + lr, T_ - 1);
    const float* Ap = Hsg + ((size_t)b * T_ + tcl) * H_ + 4 * half;
    const float* Wp = Wq + (size_t)(n0 + lr) * H_ + 4 * half;
    v8f a0 = {0,0,0,0,0,0,0,0}, a1 = {0,0,0,0,0,0,0,0};
    v8f a2 = {0,0,0,0,0,0,0,0}, a3 = {0,0,0,0,0,0,0,0};
    v4f aC  = *(const v4f*)(Ap);
    v4f b0C = *(const v4f*)(Wp);
    v4f b1C = *(const v4f*)(Wp + (size_t)16 * H_);
    v4f b2C = *(const v4f*)(Wp + (size_t)32 * H_);
    v4f b3C = *(const v4f*)(Wp + (size_t)48 * H_);
    for (int s = 0; s < (H_ / 8) - 1; ++s) {
      const int kn = (s + 1) * 8;
      v4f aN  = *(const v4f*)(Ap + kn);
      v4f b0N = *(const v4f*)(Wp + kn);
      v4f b1N = *(const v4f*)(Wp + (size_t)16 * H_ + kn);
      v4f b2N = *(const v4f*)(Wp + (size_t)32 * H_ + kn);
      v4f b3N = *(const v4f*)(Wp + (size_t)48 * H_ + kn);
      a0 = wmma4(lo2(aC), lo2(b0C), a0);
      a1 = wmma4(lo2(aC), lo2(b1C), a1);
      a2 = wmma4(lo2(aC), lo2(b2C), a2);
      a3 = wmma4(lo2(aC), lo2(b3C), a3);
      a0 = wmma4(hi2(aC), hi2(b0C), a0);
      a1 = wmma4(hi2(aC), hi2(b1C), a1);
      a2 = wmma4(hi2(aC), hi2(b2C), a2);
      a3 = wmma4(hi2(aC), hi2(b3C), a3);
      aC = aN; b0C = b0N; b1C = b1N; b2C = b2N; b3C = b3N;
    }
    a0 = wmma4(lo2(aC), lo2(b0C), a0);
    a1 = wmma4(lo2(aC), lo2(b1C), a1);
    a2 = wmma4(lo2(aC), lo2(b2C), a2);
    a3 = wmma4(lo2(aC), lo2(b3C), a3);
    a0 = wmma4(hi2(aC), hi2(b0C), a0);
    a1 = wmma4(hi2(aC), hi2(b1C), a1);
    a2 = wmma4(hi2(aC), hi2(b2C), a2);
    a3 = wmma4(hi2(aC), hi2(b3C), a3);
#pragma unroll
    for (int v = 0; v < 8; ++v) {
      int row = v + 8 * half;
#pragma unroll
      for (int j = 0; j < 4; ++j) {
        int col = n0 + j * 16 + lr;
        float s = (j == 0 ? a0[v] : j == 1 ? a1[v] : j == 2 ? a2[v] : a3[v]);
        qlds[row * HS + col] = (s + bq[col]) * scale;
      }
    }
  }
  __syncthreads();

  // phase 1: logits tile (16 x 512) over all keys n
  v8f l0 = {0,0,0,0,0,0,0,0}, l1 = {0,0,0,0,0,0,0,0};
  v8f l2 = {0,0,0,0,0,0,0,0}, l3 = {0,0,0,0,0,0,0,0};
  {
    const float* qp = &qlds[lr * HS + 4 * half];
    const float* Kp = Kmat + (size_t)b * N_ * H_ + (size_t)(n0 + lr) * H_ + 4 * half;
    v4f aC  = *(const v4f*)(qp);
    v4f b0C = *(const v4f*)(Kp);
    v4f b1C = *(const v4f*)(Kp + (size_t)16 * H_);
    v4f b2C = *(const v4f*)(Kp + (size_t)32 * H_);
    v4f b3C = *(const v4f*)(Kp + (size_t)48 * H_);
    for (int s = 0; s < (H_ / 8) - 1; ++s) {
      const int kn = (s + 1) * 8;
      v4f aN  = *(const v4f*)(qp + kn);
      v4f b0N = *(const v4f*)(Kp + kn);
      v4f b1N = *(const v4f*)(Kp + (size_t)16 * H_ + kn);
      v4f b2N = *(const v4f*)(Kp + (size_t)32 * H_ + kn);
      v4f b3N = *(const v4f*)(Kp + (size_t)48 * H_ + kn);
      l0 = wmma4(lo2(aC), lo2(b0C), l0);
      l1 = wmma4(lo2(aC), lo2(b1C), l1);
      l2 = wmma4(lo2(aC), lo2(b2C), l2);
      l3 = wmma4(lo2(aC), lo2(b3C), l3);
      l0 = wmma4(hi2(aC), hi2(b0C), l0);
      l1 = wmma4(hi2(aC), hi2(b1C), l1);
      l2 = wmma4(hi2(aC), hi2(b2C), l2);
      l3 = wmma4(hi2(aC), hi2(b3C), l3);
      aC = aN; b0C = b0N; b1C = b1N; b2C = b2N; b3C = b3N;
    }
    l0 = wmma4(lo2(aC), lo2(b0C), l0);
    l1 = wmma4(lo2(aC), lo2(b1C), l1);
    l2 = wmma4(lo2(aC), lo2(b2C), l2);
    l3 = wmma4(lo2(aC), lo2(b3C), l3);
    l0 = wmma4(hi2(aC), hi2(b0C), l0);
    l1 = wmma4(hi2(aC), hi2(b1C), l1);
    l2 = wmma4(hi2(aC), hi2(b2C), l2);
    l3 = wmma4(hi2(aC), hi2(b3C), l3);
  }
  __syncthreads();  // everyone finished reading Q before we overwrite
#pragma unroll
  for (int v = 0; v < 8; ++v) {
    int row = v + 8 * half;
#pragma unroll
    for (int j = 0; j < 4; ++j) {
      int col = n0 + j * 16 + lr;
      float s = (j == 0 ? l0[v] : j == 1 ? l1[v] : j == 2 ? l2[v] : l3[v]);
      qlds[row * HS + col] = s;  // buffer reused for logits
    }
  }
  __syncthreads();

  // phase 2: masked log-sum-exp + CE; wave handles rows wave and wave+8
  for (int rr = wave; rr < 16; rr += 8) {
    int t = t0 + rr;
    if (t >= T_) continue;  // wave-uniform
    const float* rowp = &qlds[rr * HS];
    float m = -1e30f;
    for (int n = lane; n <= t; n += 32) m = fmaxf(m, rowp[n]);
#pragma unroll
    for (int s = 16; s > 0; s >>= 1) m = fmaxf(m, __shfl_xor(m, s, 32));
    float sum = 0.0f;
    for (int n = lane; n <= t; n += 32) sum += expf(rowp[n] - m);
#pragma unroll
    for (int s = 16; s > 0; s >>= 1) sum += __shfl_xor(sum, s, 32);
    if (lane == 0) {
      int tp = parent[b * N_ + t + 1];
      if (tp >= 0 && tp <= t) {
        float ce = -(rowp[tp] - (m + logf(sum)));
        atomicAdd(&out[b * 3 + 0], ce * (1.0f / T_));
      }
    }
  }
}

// ---------------------------------------------------------------------------
// Per-(b,t): mu/logvar heads (P=3, K=2H), pos NLL and stop BCE losses.
// ---------------------------------------------------------------------------
__global__ void posstop_kernel(
    const float* __restrict__ Hs, const float* __restrict__ emb,
    const float* __restrict__ tpos, const int* __restrict__ parent,
    const float* __restrict__ Wmu, const float* __restrict__ bmu,
    const float* __restrict__ Wlv, const float* __restrict__ blv,
    const float* __restrict__ Ws, const float* __restrict__ bs,
    float* __restrict__ out) {
  int idx = blockIdx.x * blockDim.x + threadIdx.x;
  if (idx >= B_ * T_) return;
  int b = idx / T_;
  int t = idx - b * T_;
  int tp = parent[b * N_ + t + 1];
  int chosen = tp > 0 ? tp : 0;
  const float* hrow = Hs + (size_t)idx * H_;                  // row b*T+t
  const float* erow = emb + ((size_t)b * N_ + chosen) * H_;   // parent_embed
  float mu[3] = {bmu[0], bmu[1], bmu[2]};
  float lv[3] = {blv[0], blv[1], blv[2]};
  float sacc = bs[0];
  for (int k = 0; k < H_; ++k) {
    float hv = hrow[k], ev = erow[k];
#pragma unroll
    for (int p = 0; p < 3; ++p) {
      mu[p] += hv * Wmu[p * (2 * H_) + k] + ev * Wmu[p * (2 * H_) + H_ + k];
      lv[p] += hv * Wlv[p * (2 * H_) + k] + ev * Wlv[p * (2 * H_) + H_ + k];
    }
    sacc += hv * Ws[k];
  }
  float pl = 0.0f;
#pragma unroll
  for (int p = 0; p < 3; ++p) {
    float ppos = (chosen == 0) ? 0.0f
                               : tpos[((size_t)b * N_ + chosen) * P_ + p];
    float off = tpos[((size_t)b * N_ + t + 1) * P_ + p] - ppos;
    float d = off - mu[p];
    pl += 0.5f * (lv[p] + d * d / (expf(lv[p]) + 1e-8f));
  }
  atomicAdd(&out[b * 3 + 1], pl * (1.0f / T_));
  float bce = (t == T_ - 1) ? softplusf_(-sacc) : softplusf_(sacc);
  atomicAdd(&out[b * 3 + 2], bce * (1.0f / T_));
}

// ---------------------------------------------------------------------------
extern "C" void kernel_launch(void* const* d_in, const int* in_sizes, int n_in,
                              void* d_out, int out_size, void* d_ws,
                              size_t ws_size, hipStream_t stream) {
  const float* z    = (const float*)d_in[0];
  const float* tpos = (const float*)d_in[1];
  const float* Wzh  = (const float*)d_in[2];
  const float* bzh  = (const float*)d_in[3];
  const float* Wne  = (const float*)d_in[4];
  const float* bne  = (const float*)d_in[5];
  const float* Wih  = (const float*)d_in[6];
  const float* bih  = (const float*)d_in[7];
  const float* Whh  = (const float*)d_in[8];
  const float* bhh  = (const float*)d_in[9];
  const float* Wq   = (const float*)d_in[10];
  const float* bq   = (const float*)d_in[11];
  const float* Wk   = (const float*)d_in[12];
  const float* bk   = (const float*)d_in[13];
  const float* Wmu  = (const float*)d_in[14];
  const float* bmu  = (const float*)d_in[15];
  const float* Wlv  = (const float*)d_in[16];
  const float* blv  = (const float*)d_in[17];
  const float* Ws   = (const float*)d_in[18];
  const float* bs   = (const float*)d_in[19];
  const int* parent = (const int*)d_in[20];
  float* out = (float*)d_out;

  char* ws = (char*)d_ws;
  size_t off = 0;
  auto alloc = [&](size_t nfloats) {
    float* p = (float*)(ws + off);
    off += ((nfloats * sizeof(float) + 255) / 256) * 256;
    return p;
  };
  float* emb = alloc((size_t)B_ * N_ * H_);        // 134 MB
  float* Xp  = alloc((size_t)B_ * T_ * 3 * H_);    // 402 MB
  float* Hsb = alloc((size_t)B_ * T_ * H_);        // 134 MB
  float* Kb  = alloc((size_t)B_ * N_ * H_);        // 134 MB
  float* h0  = alloc((size_t)B_ * H_);
  (void)ws_size; (void)in_sizes; (void)n_in;

  hipMemsetAsync(d_out, 0, (size_t)out_size * sizeof(float), stream);

  {  // embeds
    size_t total = (size_t)B_ * N_ * H_;
    embed_kernel<<<(unsigned)((total + 255) / 256), 256, 0, stream>>>(
        tpos, Wne, bne, emb);
  }
  {  // h0 = tanh(z @ W_zh^T + b_zh)
    int waves = (B_ / 16) * (H_ / 64);
    gemm_wmma<<<(waves + 3) / 4, 128, 0, stream>>>(
        z, (long long)L_, Wzh, bzh, h0, (long long)H_, B_, H_, L_, 0, 1);
  }
  {  // Xp = embeds[:, :T] @ W_ih^T + b_ih   (rowmap 1: bt -> b*N+t)
    int waves = (B_ * T_ / 16) * (3 * H_ / 64);
    gemm_wmma<<<(waves + 3) / 4, 128, 0, stream>>>(
        emb, (long long)H_, Wih, bih, Xp, (long long)(3 * H_),
        B_ * T_, 3 * H_, H_, 1, 0);
  }
  {  // K = embeds @ W_k^T + b_k
    int waves = (B_ * N_ / 16) * (H_ / 64);
    gemm_wmma<<<(waves + 3) / 4, 128, 0, stream>>>(
        emb, (long long)H_, Wk, bk, Kb, (long long)H_, B_ * N_, H_, H_, 0, 0);
  }
  // sequential GRU over T steps (persistent, 8 workgroups x 16 batch rows)
  gru_kernel<<<B_ / 16, 512, 0, stream>>>(h0, Xp, Whh, bhh, Hsb);
  // fused Q-projection + logits + masked softmax + parent CE
  attn_kernel<<<B_ * 32, 256, 0, stream>>>(Hsb, Wq, bq, Kb, parent, out);
  // mu/logvar/pos NLL + stop BCE
  {
    int total = B_ * T_;
    posstop_kernel<<<(total + 255) / 256, 256, 0, stream>>>(
        Hsb, emb, tpos, parent, Wmu, bmu, Wlv, blv, Ws, bs, out);
  }
}